// SeparateSourceMaskingDense_74448963108907
// MI455X (gfx1250) — compile-verified
//
#include <hip/hip_runtime.h>

#define N_ROWS   131072
#define F_IN     256
#define FEAT     64
#define NSRC     32
#define F_OUT    2048      // NSRC * FEAT
#define M_TILE   64
#define PERM_LEN (N_ROWS + 2048)
#define MAX_TILES (N_ROWS / M_TILE + NSRC)   // 2080
#define KC       64        // K chunk staged in LDS (4 chunks)
#define AST      68        // LDS stride: 68 % 64 == 4 -> conflict-free, even -> b64 ok
#define BST      68

typedef float v2f __attribute__((ext_vector_type(2)));
typedef float v8f __attribute__((ext_vector_type(8)));

__global__ void init_kernel(int* __restrict__ counts, int* __restrict__ perm) {
    int idx = blockIdx.x * 256 + threadIdx.x;
    if (idx < PERM_LEN) perm[idx] = -1;
    if (idx < NSRC) counts[idx] = 0;
}

__global__ void hist_kernel(const int* __restrict__ src, int* __restrict__ counts) {
    __shared__ int h[NSRC];
    int tid = threadIdx.x;
    if (tid < NSRC) h[tid] = 0;
    __syncthreads();
    int idx = blockIdx.x * 256 + tid;
    if (idx < N_ROWS) atomicAdd(&h[src[idx]], 1);
    __syncthreads();
    if (tid < NSRC) atomicAdd(&counts[tid], h[tid]);
}

__global__ void prefix_kernel(const int* __restrict__ counts,
                              int* __restrict__ po, int* __restrict__ cursors) {
    if (threadIdx.x == 0) {
        int run = 0;
        for (int s = 0; s < NSRC; ++s) {
            po[s] = run;
            cursors[s] = run;
            run += (counts[s] + (M_TILE - 1)) & ~(M_TILE - 1);
        }
        po[NSRC] = run;
    }
}

__global__ void scatter_kernel(const int* __restrict__ src,
                               int* __restrict__ cursors, int* __restrict__ perm) {
    int idx = blockIdx.x * 256 + threadIdx.x;
    if (idx < N_ROWS) {
        int s = src[idx];
        int pos = atomicAdd(&cursors[s], 1);
        perm[pos] = idx;
    }
}

// One workgroup: 64 rows (one bucket) x 64 features, K=256 in four 64-chunks.
// 8 waves; wave = m_pair*4 + n_tile. Each wave owns TWO 16x16 tiles (m-tiles
// m_pair*2 and m_pair*2+1) sharing one B fragment: 3 b64 LDS reads feed 2
// WMMAs with independent accumulator chains.
__global__ __launch_bounds__(256)
void gemm_kernel(const float* __restrict__ x, const float* __restrict__ W,
                 const float* __restrict__ b, const int* __restrict__ perm,
                 const int* __restrict__ po, float* __restrict__ out) {
    __shared__ float As[M_TILE * AST];   // 64 rows x 64 K (gathered)
    __shared__ float Bs[FEAT * BST];     // 64 cols x 64 K (transposed W block)
    __shared__ int   rowsS[M_TILE];
    __shared__ float biasS[FEAT];
    __shared__ int   poS[NSRC + 1];

    const int tid = threadIdx.x;
    if (tid < NSRC + 1) poS[tid] = po[tid];
    __syncthreads();

    const int rowbase = blockIdx.x * M_TILE;
    if (rowbase >= poS[NSRC]) return;            // uniform per block

    int s = 0;                                    // uniform: find bucket
    while (s < NSRC - 1 && rowbase >= poS[s + 1]) ++s;

    if (tid < M_TILE) rowsS[tid] = perm[rowbase + tid];
    if (tid < FEAT)   biasS[tid] = b[s * FEAT + tid];
    __syncthreads();

    const int wave = tid >> 5, lane = tid & 31;
    const int m_pair = wave >> 2, n_tile = wave & 3;
    const int l15 = lane & 15;
    const int kh  = (lane >> 4) << 1;             // lanes 16-31 hold K+2,K+3
    const float* Ap0 = &As[(m_pair * 32 +      l15) * AST + kh];
    const float* Ap1 = &As[(m_pair * 32 + 16 + l15) * AST + kh];
    const float* Bp  = &Bs[(n_tile * 16 +      l15) * BST + kh];

    v8f acc0 = {0.f, 0.f, 0.f, 0.f, 0.f, 0.f, 0.f, 0.f};
    v8f acc1 = {0.f, 0.f, 0.f, 0.f, 0.f, 0.f, 0.f, 0.f};

    for (int c = 0; c < F_IN / KC; ++c) {
        __syncthreads();                          // protect LDS from prev chunk readers
        // ---- load A chunk: 64 rows x 64 floats, gathered via perm; float4
        {
            int k4 = tid & 15;                    // 16 float4 per 64-float row
            int rs = tid >> 4;
            #pragma unroll
            for (int it = 0; it < 4; ++it) {
                int r = it * 16 + rs;
                int row = rowsS[r];
                float4 v = make_float4(0.f, 0.f, 0.f, 0.f);
                if (row >= 0)
                    v = ((const float4*)x)[(size_t)row * (F_IN / 4) + c * (KC / 4) + k4];
                *(float4*)&As[r * AST + k4 * 4] = v;
            }
        }
        // ---- load B chunk transposed: Bs[j*BST + k] = W[(c*KC+k)*F_OUT + s*64 + j]
        {
            int j = tid & 63;
            int ks = tid >> 6;
            const float* Wp = W + (size_t)s * FEAT + j;
            #pragma unroll 4
            for (int it = 0; it < KC / 4; ++it) {
                int k = it * 4 + ks;
                Bs[j * BST + k] = Wp[(size_t)(c * KC + k) * F_OUT];
            }
        }
        __syncthreads();
        // ---- 16 k-steps: one B fragment feeds two WMMAs (independent chains)
        #pragma unroll 8
        for (int kk = 0; kk < KC / 4; ++kk) {
            v2f bb = *(const v2f*)(Bp  + kk * 4);
            v2f a0 = *(const v2f*)(Ap0 + kk * 4);
            v2f a1 = *(const v2f*)(Ap1 + kk * 4);
            acc0 = __builtin_amdgcn_wmma_f32_16x16x4_f32(
                false, a0, false, bb, (short)0, acc0, false, false);
            acc1 = __builtin_amdgcn_wmma_f32_16x16x4_f32(
                false, a1, false, bb, (short)0, acc1, false, false);
        }
    }

    // ---- store: acc[r] is (M = r + (lane>=16)*8, N = lane&15) of each tile
    const int col  = n_tile * 16 + l15;
    const float bias = biasS[col];
    const int mb0 = m_pair * 32 +      ((lane >> 4) << 3);
    const int mb1 = m_pair * 32 + 16 + ((lane >> 4) << 3);
    #pragma unroll
    for (int r = 0; r < 8; ++r) {
        int g0 = rowsS[mb0 + r];
        if (g0 >= 0) out[(size_t)g0 * FEAT + col] = acc0[r] + bias;
        int g1 = rowsS[mb1 + r];
        if (g1 >= 0) out[(size_t)g1 * FEAT + col] = acc1[r] + bias;
    }
}

extern "C" void kernel_launch(void* const* d_in, const int* in_sizes, int n_in,
                              void* d_out, int out_size, void* d_ws, size_t ws_size,
                              hipStream_t stream) {
    const float* x      = (const float*)d_in[0];
    const int*   source = (const int*)d_in[1];
    const float* W      = (const float*)d_in[2];
    const float* b      = (const float*)d_in[3];
    float*       out    = (float*)d_out;

    int* counts  = (int*)d_ws;        // 32
    int* po      = counts + 32;       // 33
    int* cursors = counts + 72;       // 32
    int* perm    = counts + 128;      // PERM_LEN

    init_kernel<<<(PERM_LEN + 255) / 256, 256, 0, stream>>>(counts, perm);
    hist_kernel<<<N_ROWS / 256, 256, 0, stream>>>(source, counts);
    prefix_kernel<<<1, 32, 0, stream>>>(counts, po, cursors);
    scatter_kernel<<<N_ROWS / 256, 256, 0, stream>>>(source, cursors, perm);
    gemm_kernel<<<MAX_TILES, 256, 0, stream>>>(x, W, b, perm, po, out);
}